// SetAbstraction_59682865545240
// MI455X (gfx1250) — compile-verified
//
#include <hip/hip_runtime.h>
#include <hip/hip_bf16.h>

// ---------------- problem constants (from reference) ----------------
#define NPTS        50000
#define MSUB        2048
#define KOUT        64
#define RADIUS2     0.04f         // 0.2^2 : filter/sort in squared-distance domain

// ---------------- FPS kernel config ----------------
#define FPS_THREADS 1024
#define PPT         49            // ceil(50000/1024)

// ---------------- grouping kernel config ----------------
#define CPB         16            // centroids per block (one WMMA M-tile)
#define TILE_V      128           // vertices per block iteration (8 waves x 16)
#define CAP         2048          // per-centroid LDS candidate capacity (E[count]~1675)

typedef __attribute__((ext_vector_type(2))) float v2f;
typedef __attribute__((ext_vector_type(8))) float v8f;

// =====================================================================
// Kernel 1: farthest point sampling. One persistent workgroup, 32 waves.
// min-distance array lives in VGPRs (49/lane); vertex coords streamed
// from the 192MB L2 every step (600KB working set, L2-resident).
// =====================================================================
__global__ void __launch_bounds__(FPS_THREADS)
fps_kernel(const float* __restrict__ verts, float4* __restrict__ cents)
{
    const int tid  = threadIdx.x;
    const int lane = tid & 31;
    const int wv   = tid >> 5;

    __shared__ unsigned long long wred[FPS_THREADS / 32];
    __shared__ float4 curb;

    float md[PPT];
#pragma unroll
    for (int i = 0; i < PPT; ++i) md[i] = 3.402823466e38f;

    if (tid == 0) {
        float4 c0 = make_float4(verts[0], verts[1], verts[2], 0.f);
        curb = c0;
        cents[0] = c0;                       // idx[0] == 0 in the reference scan
    }
    __syncthreads();

    for (int t = 1; t < MSUB; ++t) {
        const float cx = curb.x, cy = curb.y, cz = curb.z;

        unsigned long long bk = 0ull;        // (md_bits<<32) | (~idx) -> argmax, tie->lowest idx
#pragma unroll
        for (int i = 0; i < PPT; ++i) {
            const int p = tid + i * FPS_THREADS;
            if (p < NPTS) {
                const float dx = verts[3 * p + 0] - cx;
                const float dy = verts[3 * p + 1] - cy;
                const float dz = verts[3 * p + 2] - cz;
                const float d  = dx * dx + dy * dy + dz * dz;
                md[i] = fminf(md[i], d);
                const unsigned long long key =
                    ((unsigned long long)__float_as_uint(md[i]) << 32) |
                    (unsigned long long)(0xFFFFFFFFu - (unsigned)p);
                bk = key > bk ? key : bk;
            }
        }
        // intra-wave max reduce (wave32)
#pragma unroll
        for (int off = 16; off; off >>= 1) {
            const unsigned long long ok = __shfl_down(bk, off, 32);
            bk = ok > bk ? ok : bk;
        }
        if (lane == 0) wred[wv] = bk;
        __syncthreads();
        if (tid < 32) {
            unsigned long long k2 = wred[tid];
#pragma unroll
            for (int off = 16; off; off >>= 1) {
                const unsigned long long ok = __shfl_down(k2, off, 32);
                k2 = ok > k2 ? ok : k2;
            }
            if (tid == 0) {
                const unsigned w = 0xFFFFFFFFu - (unsigned)(k2 & 0xFFFFFFFFull);
                const float4 c4 = make_float4(verts[3 * w + 0], verts[3 * w + 1],
                                              verts[3 * w + 2], 0.f);
                curb = c4;
                cents[t] = c4;
            }
        }
        __syncthreads();
    }
}

// =====================================================================
// Kernel 2: fused dist^2 (WMMA f32 16x16x4, K=3 padded) + radius filter
// into 320KB-class WGP LDS candidate lists + per-centroid top-64 select.
// All comparisons done on squared distances (sqrt is monotonic and the
// distance value itself never reaches the output) -> no sqrt in the hot
// loop. Grid: 128 blocks x 256 threads.
// =====================================================================
__global__ void __launch_bounds__(256)
group_kernel(const float* __restrict__ verts,
             const float4* __restrict__ cents,
             float* __restrict__ out)
{
    extern __shared__ unsigned long long dsm[];
    unsigned long long* buf  = dsm;                               // CPB*CAP keys
    float* stage = (float*)(dsm + CPB * CAP);                     // TILE_V*4 floats (x,y,z,|v|^2)
    float* cb    = stage + TILE_V * 4;                            // CPB*4 centroid coords
    int*   cnt   = (int*)(cb + CPB * 4);                          // CPB counters

    const int tid  = threadIdx.x;
    const int lane = tid & 31;
    const int wave = tid >> 5;
    const int cbase = blockIdx.x * CPB;

    if (tid < CPB) {
        const float4 c = cents[cbase + tid];
        cb[tid * 4 + 0] = c.x; cb[tid * 4 + 1] = c.y;
        cb[tid * 4 + 2] = c.z; cb[tid * 4 + 3] = 0.f;
        cnt[tid] = 0;
    }
    __syncthreads();

    // ---- A matrix (16x4 f32; lanes 0-15 hold K0/K1, lanes 16-31 K2/K3).
    // Non-divergent: one b64-wide LDS read at a lane-computed offset, then
    // zero the K3 pad on the upper half with a single select.
    const int ml   = lane & 15;
    const int hsel = (lane & 16) >> 3;            // 0 for lanes 0-15, 2 for 16-31
    v2f a;
    a.x = cb[ml * 4 + hsel + 0];                  // K0 (x) or K2 (z)
    a.y = (lane < 16) ? cb[ml * 4 + 1] : 0.f;     // K1 (y) or K3 (pad 0)

    for (int base = 0; base < NPTS; base += TILE_V) {
        // ---- stage 128 vertices + squared norms into LDS
        if (tid < TILE_V) {
            const int p = base + tid;
            float x = 0.f, y = 0.f, z = 0.f, sq = 3.402823466e38f;
            if (p < NPTS) {
                x = verts[3 * p + 0]; y = verts[3 * p + 1]; z = verts[3 * p + 2];
                sq = x * x + y * y + z * z;
                __builtin_prefetch(verts + 3 * (p + TILE_V), 0, 1);   // global_prefetch_b8
            }
            stage[tid * 4 + 0] = x; stage[tid * 4 + 1] = y;
            stage[tid * 4 + 2] = z; stage[tid * 4 + 3] = sq;
        }
        __syncthreads();

        // ---- B matrix (4x16 f32) for this wave's 16-vertex tile (same trick)
        const int vt = wave * 16 + ml;
        v2f b;
        b.x = stage[vt * 4 + hsel + 0];
        b.y = (lane < 16) ? stage[vt * 4 + 1] : 0.f;
        const float sq = stage[vt * 4 + 3];
        const int   n  = base + vt;

        // ---- 16x16 gram tile via CDNA5 WMMA
        float g8[8];
#if __has_builtin(__builtin_amdgcn_wmma_f32_16x16x4_f32)
        {
            v8f c8 = {};
            v8f d = __builtin_amdgcn_wmma_f32_16x16x4_f32(
                false, a, false, b, (short)0, c8, false, false);
#pragma unroll
            for (int r = 0; r < 8; ++r) g8[r] = d[r];
        }
#else
        {
            const float bx = stage[vt * 4 + 0], by = stage[vt * 4 + 1], bz = stage[vt * 4 + 2];
#pragma unroll
            for (int r = 0; r < 8; ++r) {
                const int m = r + ((lane & 16) >> 1);
                g8[r] = cb[m * 4 + 0] * bx + cb[m * 4 + 1] * by + cb[m * 4 + 2] * bz;
            }
        }
#endif
        // ---- radius filter in squared domain -> LDS candidate append
#pragma unroll
        for (int r = 0; r < 8; ++r) {
            const int   m  = r + ((lane & 16) >> 1);     // D layout: lanes>=16 = rows 8..15
            const float d2 = fabsf(2.f * sq - 2.f * g8[r]);   // bug-faithful |2|v|^2 - 2 c.v|
            if (n < NPTS && d2 <= RADIUS2) {
                const int pos = atomicAdd(&cnt[m], 1);
                if (pos < CAP)
                    buf[m * CAP + pos] =
                        ((unsigned long long)__float_as_uint(d2) << 32) |
                        (unsigned long long)(unsigned)n;
            }
        }
        __syncthreads();
    }

    // ---- per-centroid top-64 ascending (d2 order == dist order; ties by idx)
    volatile unsigned long long* vbuf = buf;
    float* out_nb   = out;
    float* out_mask = out + (size_t)MSUB * KOUT * 3;

    for (int m = wave; m < CPB; m += 8) {        // 8 waves -> 2 centroids each
        const int c     = cbase + m;
        int total = cnt[m]; if (total > CAP) total = CAP;
        const int nsel  = total < KOUT ? total : KOUT;

        for (int k = 0; k < KOUT; ++k) {
            if (k < nsel) {
                unsigned long long bkey = ~0ull; int bslot = -1;
                for (int s = lane; s < total; s += 32) {
                    const unsigned long long v = vbuf[m * CAP + s];
                    if (v < bkey) { bkey = v; bslot = s; }
                }
#pragma unroll
                for (int off = 16; off; off >>= 1) {
                    const unsigned long long ok = __shfl_down(bkey, off, 32);
                    const int os = __shfl_down(bslot, off, 32);
                    if (ok < bkey) { bkey = ok; bslot = os; }
                }
                if (lane == 0) {
                    const unsigned idx = (unsigned)(bkey & 0xFFFFFFFFull);
                    vbuf[m * CAP + bslot] = ~0ull;           // remove winner
                    const size_t o = ((size_t)c * KOUT + k) * 3;
                    out_nb[o + 0] = verts[3 * idx + 0];
                    out_nb[o + 1] = verts[3 * idx + 1];
                    out_nb[o + 2] = verts[3 * idx + 2];
                    out_mask[(size_t)c * KOUT + k] = 1.f;
                }
            } else if (lane == 0) {
                const size_t o = ((size_t)c * KOUT + k) * 3;
                out_nb[o + 0] = 0.f; out_nb[o + 1] = 0.f; out_nb[o + 2] = 0.f;
                out_mask[(size_t)c * KOUT + k] = 0.f;
            }
        }
    }
}

// =====================================================================
extern "C" void kernel_launch(void* const* d_in, const int* in_sizes, int n_in,
                              void* d_out, int out_size, void* d_ws, size_t ws_size,
                              hipStream_t stream)
{
    (void)in_sizes; (void)n_in; (void)out_size; (void)ws_size;
    // d_in[0] = vertex_features [50000,64]  (unused by the reference output)
    // d_in[1] = vertices        [50000,3]
    const float* verts = (const float*)d_in[1];
    float4* cents = (float4*)d_ws;                 // 2048 * 16B = 32KB scratch

    fps_kernel<<<1, FPS_THREADS, 0, stream>>>(verts, cents);

    const size_t shmem = (size_t)CPB * CAP * sizeof(unsigned long long)   // 256KB keys
                       + (size_t)(TILE_V * 4 + CPB * 4) * sizeof(float)   // staging + centroids
                       + (size_t)CPB * sizeof(int);                       // counters
    group_kernel<<<MSUB / CPB, 256, shmem, stream>>>(verts, cents, (float*)d_out);
}